// GELU208_39857296507265
// MI455X (gfx1250) — compile-verified
//
#include <hip/hip_runtime.h>
#include <math.h>

#define DCOL 2048
#define TPB  256
#define EPT  8      // DCOL / TPB
#define KSEL 16
#define NWAVE (TPB/32)
#define LCAP 256    // candidate list capacity (boundary bin rarely > ~150)

typedef float v4f __attribute__((ext_vector_type(4)));
typedef unsigned long long u64;

// ---- wave32 butterfly reductions via ds_swizzle (full 32-lane XOR) --------
template<int IMM>
__device__ __forceinline__ u64 swz64(u64 v) {
    unsigned lo = (unsigned)__builtin_amdgcn_ds_swizzle((int)(unsigned)(v & 0xffffffffull), IMM);
    unsigned hi = (unsigned)__builtin_amdgcn_ds_swizzle((int)(unsigned)(v >> 32), IMM);
    return ((u64)hi << 32) | lo;
}
__device__ __forceinline__ u64 wave_max_u64(u64 v) {
    u64 o;
    o = swz64<0x041f>(v); v = o > v ? o : v;   // xor 1
    o = swz64<0x081f>(v); v = o > v ? o : v;   // xor 2
    o = swz64<0x101f>(v); v = o > v ? o : v;   // xor 4
    o = swz64<0x201f>(v); v = o > v ? o : v;   // xor 8
    o = swz64<0x401f>(v); v = o > v ? o : v;   // xor 16
    return v;
}
template<int IMM>
__device__ __forceinline__ float swzf(float v) {
    return __int_as_float(__builtin_amdgcn_ds_swizzle(__float_as_int(v), IMM));
}
__device__ __forceinline__ float wave_sum_f(float v) {
    v += swzf<0x041f>(v); v += swzf<0x081f>(v); v += swzf<0x101f>(v);
    v += swzf<0x201f>(v); v += swzf<0x401f>(v);
    return v;
}

__device__ __forceinline__ float gate_of(float z, float beta, float gamma) {
    float g = 1.0f + beta * tanhf(gamma * z);
    return fminf(fmaxf(g, 0.1f), 8.0f);
}

// ---------------------------------------------------------------------------
// Prep: ||ema_out||, tau = exp(log_tau), beta/gamma = softplus(...)
// ws[0]=tau ws[1]=beta ws[2]=gamma ws[3]=max(||ema_out||,1e-12)
// ---------------------------------------------------------------------------
__global__ __launch_bounds__(TPB) void topk_gate_prep(
    const float* __restrict__ ema_out,
    const float* __restrict__ log_tau,
    const float* __restrict__ log_beta,
    const float* __restrict__ log_gamma,
    float* __restrict__ ws, int D)
{
    __shared__ float red[NWAVE];
    const int tid = threadIdx.x;
    float s = 0.f;
    for (int d = tid; d < D; d += TPB) { float v = ema_out[d]; s += v * v; }
    s = wave_sum_f(s);
    if ((tid & 31) == 0) red[tid >> 5] = s;
    __syncthreads();
    if (tid == 0) {
        float t = 0.f;
        for (int w = 0; w < NWAVE; ++w) t += red[w];
        ws[3] = fmaxf(sqrtf(t), 1e-12f);
        ws[0] = expf(log_tau[0]);
        ws[1] = log1pf(expf(log_beta[0]));   // softplus
        ws[2] = log1pf(expf(log_gamma[0]));  // softplus
    }
}

// ---------------------------------------------------------------------------
// Main fused kernel: one block per token row.
// ---------------------------------------------------------------------------
__global__ __launch_bounds__(TPB) void topk_gate_fused(
    const float* __restrict__ x,
    const float* __restrict__ ema_mean,
    const float* __restrict__ ema_sq,
    const float* __restrict__ ema_out,
    const float* __restrict__ ws,
    float* __restrict__ out)
{
    __shared__ float s_x[DCOL];      // staged input row (8 KB)
    __shared__ float s_gate[DCOL];   // per-element gates (8 KB)
    __shared__ int   s_hist[256];    // exponent-byte histogram
    __shared__ u64   s_list[LCAP];   // boundary-bin candidates
    __shared__ float s_red[2 * NWAVE];
    __shared__ float s_tot[2];
    __shared__ int   s_thr[2];       // e*, r
    __shared__ int   s_cnt;

    const int tid  = threadIdx.x;
    const int lane = tid & 31;
    const int wv   = tid >> 5;
    const long long row = blockIdx.x;
    const float* xrow = x + row * (long long)DCOL;

    const float tau   = ws[0];
    const float beta  = ws[1];
    const float gamma = ws[2];
    const float enorm = ws[3];

    // ---- async global -> LDS stage of this row (ASYNCcnt path) ------------
    {
        unsigned lds_base = (unsigned)(unsigned long long)(void*)&s_x[0];
        unsigned ldsd = lds_base + (unsigned)tid * 16u;
        unsigned voff = (unsigned)tid * 16u;
        asm volatile(
            "global_load_async_to_lds_b128 %0, %1, %2\n\t"
            "global_load_async_to_lds_b128 %0, %1, %2 offset:4096"
            :: "v"(ldsd), "v"(voff), "s"(xrow) : "memory");
    }

    const int dbase = tid * EPT;
#pragma unroll
    for (int j = 0; j < EPT; ++j) s_gate[dbase + j] = 1.0f;
    s_hist[tid] = 0;
    if (tid == 0) s_cnt = 0;

    asm volatile("s_wait_asynccnt 0" ::: "memory");
    __syncthreads();

    // ---- elementwise pass --------------------------------------------------
    float xs[EPT];
    {
        const v4f* xv4 = (const v4f*)(s_x + dbase);
        v4f a = xv4[0], b = xv4[1];
        xs[0]=a.x; xs[1]=a.y; xs[2]=a.z; xs[3]=a.w;
        xs[4]=b.x; xs[5]=b.y; xs[6]=b.z; xs[7]=b.w;
    }
    float mus[EPT], sqs[EPT], eos[EPT];
    {
        const v4f* m4 = (const v4f*)(ema_mean + dbase);
        const v4f* q4 = (const v4f*)(ema_sq   + dbase);
        const v4f* e4 = (const v4f*)(ema_out  + dbase);
        v4f m0=m4[0], m1=m4[1], q0=q4[0], q1=q4[1], e0=e4[0], e1=e4[1];
        mus[0]=m0.x; mus[1]=m0.y; mus[2]=m0.z; mus[3]=m0.w;
        mus[4]=m1.x; mus[5]=m1.y; mus[6]=m1.z; mus[7]=m1.w;
        sqs[0]=q0.x; sqs[1]=q0.y; sqs[2]=q0.z; sqs[3]=q0.w;
        sqs[4]=q1.x; sqs[5]=q1.y; sqs[6]=q1.z; sqs[7]=q1.w;
        eos[0]=e0.x; eos[1]=e0.y; eos[2]=e0.z; eos[3]=e0.w;
        eos[4]=e1.x; eos[5]=e1.y; eos[6]=e1.z; eos[7]=e1.w;
    }

    float ov[EPT], zz[EPT];
    unsigned ub[EPT];
    float ss = 0.f, dt = 0.f;
#pragma unroll
    for (int j = 0; j < EPT; ++j) {
        const float xv  = xs[j];
        const float sig = 1.0f / (1.0f + __expf(-xv));
        const float o   = xv * sig;
        ov[j] = o;
        const float mu  = mus[j];
        const float var = fmaxf(sqs[j] - mu * mu, 1e-4f);
        const float z   = (xv - mu) * __frcp_rn(sqrtf(var) + 1e-5f);
        zz[j] = z;
        ub[j] = __float_as_uint(fabsf(z));
        ss += o * o;
        dt += o * eos[j];
        atomicAdd(&s_hist[ub[j] >> 23], 1);
    }

    ss = wave_sum_f(ss);
    dt = wave_sum_f(dt);
    if (lane == 0) { s_red[wv] = ss; s_red[NWAVE + wv] = dt; }
    __syncthreads();

    // ---- wave 0: totals + histogram suffix scan -> (e*, r) -----------------
    if (wv == 0) {
        float tss = (lane < NWAVE) ? s_red[lane]         : 0.f;
        float tdt = (lane < NWAVE) ? s_red[NWAVE + lane] : 0.f;
        tss = wave_sum_f(tss);
        tdt = wave_sum_f(tdt);
        if (lane == 0) { s_tot[0] = tss; s_tot[1] = tdt; }

        int h[8]; int s = 0;
#pragma unroll
        for (int k = 0; k < 8; ++k) { h[k] = s_hist[8 * lane + k]; s += h[k]; }
        int t = s;                               // inclusive suffix over lanes
#pragma unroll
        for (int off = 1; off < 32; off <<= 1) {
            int v = __shfl_down(t, off);
            if (lane + off < 32) t += v;
        }
        const int above_lane = t - s;            // count in bins >= 8*(lane+1)
        if (above_lane < KSEL && KSEL <= t) {    // boundary is in this lane
            int cum = above_lane;
#pragma unroll
            for (int k = 7; k >= 0; --k) {       // high bin -> low bin
                int c = h[k];
                if (cum < KSEL && KSEL <= cum + c) {
                    s_thr[0] = 8 * lane + k;     // e*
                    s_thr[1] = KSEL - cum;       // r needed from bin e*
                }
                cum += c;
            }
        }
    }
    __syncthreads();

    const unsigned estar = (unsigned)s_thr[0];
    const int      rneed = s_thr[1];

    // ---- definite winners inline; boundary bin -> candidate list -----------
#pragma unroll
    for (int j = 0; j < EPT; ++j) {
        const unsigned b = ub[j] >> 23;
        if (b > estar) {
            s_gate[dbase + j] = gate_of(zz[j], beta, gamma);
        } else if (b == estar) {
            int idx = atomicAdd(&s_cnt, 1);
            if (idx < LCAP) {
                s_list[idx] = ((u64)ub[j] << 32) |
                              (u64)(0xFFFFFFFFu - (unsigned)(dbase + j));
            }
        }
    }
    __syncthreads();

    // ---- wave 0: pick top-r of boundary candidates, scatter gates ----------
    if (wv == 0) {
        int n = s_cnt; n = n < LCAP ? n : LCAP;
        u64 c[LCAP / 32];
#pragma unroll
        for (int j = 0; j < LCAP / 32; ++j) {
            const int i = lane + 32 * j;
            c[j] = (i < n) ? s_list[i] : 0ull;
        }
        u64 win = 0ull;
        for (int it = 0; it < rneed; ++it) {
            u64 best = c[0]; int bs = 0;
#pragma unroll
            for (int j = 1; j < LCAP / 32; ++j) { if (c[j] > best) { best = c[j]; bs = j; } }
            u64 wm = wave_max_u64(best);
            if (best == wm) c[bs] = 0ull;        // unique keys -> single owner
            if (lane == it) win = wm;
        }
        if (lane < rneed && win) {
            const unsigned d = 0xFFFFFFFFu - (unsigned)(win & 0xffffffffull);
            if (d < DCOL) {
                const float xv  = s_x[d];
                const float mu  = ema_mean[d];
                const float var = fmaxf(ema_sq[d] - mu * mu, 1e-4f);
                const float z   = (xv - mu) * __frcp_rn(sqrtf(var) + 1e-5f);
                s_gate[d] = gate_of(z, beta, gamma);
            }
        }
    }
    __syncthreads();

    // ---- cosine gate + final write (non-temporal) --------------------------
    const float tss = s_tot[0];
    const float tdt = s_tot[1];
    const float on  = fmaxf(sqrtf(tss), 1e-12f);
    float cs = tdt * __frcp_rn(on * enorm);
    cs = fminf(fmaxf(cs, -1.0f), 1.0f);
    const float gcos = __expf(-tau * cs);

    float gs[EPT];
    {
        const v4f* g4 = (const v4f*)(s_gate + dbase);
        v4f g0 = g4[0], g1 = g4[1];
        gs[0]=g0.x; gs[1]=g0.y; gs[2]=g0.z; gs[3]=g0.w;
        gs[4]=g1.x; gs[5]=g1.y; gs[6]=g1.z; gs[7]=g1.w;
    }
    float res[EPT];
#pragma unroll
    for (int j = 0; j < EPT; ++j) res[j] = ov[j] * gs[j] * gcos;

    v4f* op = (v4f*)(out + row * (long long)DCOL + dbase);
    v4f w0 = { res[0], res[1], res[2], res[3] };
    v4f w1 = { res[4], res[5], res[6], res[7] };
    __builtin_nontemporal_store(w0, op);
    __builtin_nontemporal_store(w1, op + 1);
}

// ---------------------------------------------------------------------------
extern "C" void kernel_launch(void* const* d_in, const int* in_sizes, int n_in,
                              void* d_out, int out_size, void* d_ws, size_t ws_size,
                              hipStream_t stream)
{
    const float* x         = (const float*)d_in[0];
    const float* ema_mean  = (const float*)d_in[1];
    const float* ema_sq    = (const float*)d_in[2];
    const float* ema_out   = (const float*)d_in[3];
    // d_in[4] = logit_decay (unused by the forward output)
    const float* log_tau   = (const float*)d_in[5];
    const float* log_beta  = (const float*)d_in[6];
    const float* log_gamma = (const float*)d_in[7];
    float* out = (float*)d_out;
    float* ws  = (float*)d_ws;

    const int D    = in_sizes[1];          // 2048
    const int rows = in_sizes[0] / D;      // B*T = 16384

    topk_gate_prep<<<1, TPB, 0, stream>>>(ema_out, log_tau, log_beta, log_gamma, ws, D);
    topk_gate_fused<<<rows, TPB, 0, stream>>>(x, ema_mean, ema_sq, ema_out, ws, out);
}